// JITWrapper_26517128085848
// MI455X (gfx1250) — compile-verified
//
#include <hip/hip_runtime.h>
#include <stdint.h>

#define SCORE_THRESH 0.5f
#define NMS_THRESH   0.5f

#define N_BOX    512
#define ROW_F4   65536            // 512*512 floats / 4 per mask row
#define MASK_OFF 3072             // floats in d_out before masks (2048 boxes + 512 labels + 512 scores)
#define KEEP_OFF (MASK_OFF + 512*512*512)

typedef float f32x4 __attribute__((ext_vector_type(4)));   // native vector for NT stores

// ---------------------------------------------------------------------------
// Kernel 1: sort by score (bitonic, desc), greedy NMS, small outputs.
// One workgroup of 512 threads (16 wave32s). All O(N^2) work is trivial.
// ---------------------------------------------------------------------------
__global__ __launch_bounds__(512) void nms_sort_kernel(
    const float* __restrict__ boxes, const float* __restrict__ scores,
    const int* __restrict__ labels, float* __restrict__ out,
    int* __restrict__ ws_order, float* __restrict__ ws_keep)
{
    __shared__ float s_score[N_BOX];
    __shared__ int   s_idx[N_BOX];
    __shared__ float s_x1[N_BOX], s_y1[N_BOX], s_x2[N_BOX], s_y2[N_BOX];
    __shared__ float s_area[N_BOX];
    __shared__ float s_keep[N_BOX];

    const int tid = threadIdx.x;
    s_score[tid] = scores[tid];
    s_idx[tid]   = tid;
    __syncthreads();

    // Bitonic sort, descending by score, carrying original index.
    for (int k = 2; k <= N_BOX; k <<= 1) {
        for (int j = k >> 1; j > 0; j >>= 1) {
            int p = tid ^ j;
            if (p > tid) {
                float a = s_score[tid], b = s_score[p];
                bool desc = ((tid & k) == 0);
                if (desc ? (a < b) : (a > b)) {
                    s_score[tid] = b; s_score[p] = a;
                    int ia = s_idx[tid]; s_idx[tid] = s_idx[p]; s_idx[p] = ia;
                }
            }
            __syncthreads();
        }
    }

    const int oi = s_idx[tid];
    const f32x4 bx = reinterpret_cast<const f32x4*>(boxes)[oi];
    s_x1[tid] = bx.x; s_y1[tid] = bx.y; s_x2[tid] = bx.z; s_y2[tid] = bx.w;
    s_area[tid] = (bx.z - bx.x) * (bx.w - bx.y);
    const float sc = s_score[tid];
    s_keep[tid] = (sc > SCORE_THRESH) ? 1.0f : 0.0f;
    __syncthreads();

    // Greedy forward suppression: equivalent to the reference fori_loop,
    // since keep[j] for j<i is final when row i is tested.
    for (int i = 0; i < N_BOX; ++i) {
        if (s_keep[i] != 0.0f && tid > i) {
            float xx1 = fmaxf(s_x1[i], s_x1[tid]);
            float yy1 = fmaxf(s_y1[i], s_y1[tid]);
            float xx2 = fminf(s_x2[i], s_x2[tid]);
            float yy2 = fminf(s_y2[i], s_y2[tid]);
            float inter = fmaxf(xx2 - xx1, 0.0f) * fmaxf(yy2 - yy1, 0.0f);
            float iou = inter / (s_area[i] + s_area[tid] - inter);
            if (iou > NMS_THRESH) s_keep[tid] = 0.0f;
        }
        __syncthreads();
    }

    const float kf = s_keep[tid];
    // boxes_out (sorted boxes * keep)
    out[tid * 4 + 0] = bx.x * kf;
    out[tid * 4 + 1] = bx.y * kf;
    out[tid * 4 + 2] = bx.z * kf;
    out[tid * 4 + 3] = bx.w * kf;
    // labels_out, scores_out, keep
    out[2048 + tid]     = (float)labels[oi] * kf;
    out[2560 + tid]     = sc * kf;
    out[KEEP_OFF + tid] = kf;
    // handoff to the mask kernel
    ws_order[tid] = oi;
    ws_keep[tid]  = kf;
}

// ---------------------------------------------------------------------------
// Kernel 2: masks_out[row] = keep[row] ? masks[order[row]] : 0
// Kept rows use the CDNA5 async global->LDS->global DMA path (ASYNCcnt),
// bypassing VGPR staging, with non-temporal hints so the 1 GiB one-shot
// stream does not thrash the 192 MB L2. Suppressed rows are zero-filled
// with non-temporal b128 stores (write-only: half the traffic).
// grid = (32, 512): 512 rows x 32 blocks; each block moves 32 KiB.
// ---------------------------------------------------------------------------
#define KCH 8   // float4 chunks per lane per phase -> 256 thr * 8 * 16B = 32 KiB LDS

__global__ __launch_bounds__(256) void mask_gather_kernel(
    const float* __restrict__ masks, float* __restrict__ out_masks,
    const int* __restrict__ ws_order, const float* __restrict__ ws_keep)
{
    __shared__ __attribute__((aligned(16))) unsigned char lbuf[256 * KCH * 16];

    const int row = blockIdx.y;
    const int t   = threadIdx.x;
    const float kf = ws_keep[row];                  // uniform per block
    const uint64_t intra = (uint64_t)blockIdx.x * (256u * KCH) + (uint64_t)t; // float4 idx (k stride 256)

    if (kf != 0.0f) {
        const int srow = ws_order[row];
        uint64_t gsrc = (uint64_t)(uintptr_t)masks +
                        (((uint64_t)srow * ROW_F4 + intra) << 4);
        uint64_t gdst = (uint64_t)(uintptr_t)out_masks +
                        (((uint64_t)row * ROW_F4 + intra) << 4);
        // Low 32 bits of a flat pointer to LDS == LDS byte offset
        // (shared aperture lives in the upper 32 bits).
        uint32_t lds0 = (uint32_t)(uintptr_t)(&lbuf[0]) + (uint32_t)t * 16u;

        #pragma unroll
        for (int k = 0; k < KCH; ++k) {
            uint32_t la = lds0 + (uint32_t)k * 4096u;     // (k*256 + t)*16
            uint64_t ga = gsrc + (uint64_t)k * 4096u;     // chunk stride 256 float4
            asm volatile("global_load_async_to_lds_b128 %0, %1, off th:TH_LOAD_NT"
                         :: "v"(la), "v"(ga) : "memory");
        }
        asm volatile("s_wait_asynccnt 0x0" ::: "memory");  // LDS now holds the tile
        #pragma unroll
        for (int k = 0; k < KCH; ++k) {
            uint32_t la = lds0 + (uint32_t)k * 4096u;
            uint64_t ga = gdst + (uint64_t)k * 4096u;
            asm volatile("global_store_async_from_lds_b128 %0, %1, off th:TH_STORE_NT"
                         :: "v"(ga), "v"(la) : "memory");
        }
        asm volatile("s_wait_asynccnt 0x0" ::: "memory");
    } else {
        const f32x4 z = {0.0f, 0.0f, 0.0f, 0.0f};
        f32x4* dst = reinterpret_cast<f32x4*>(out_masks) +
                     (uint64_t)row * ROW_F4 + intra;
        #pragma unroll
        for (int k = 0; k < KCH; ++k)
            __builtin_nontemporal_store(z, &dst[(uint64_t)k * 256u]);
    }
}

// ---------------------------------------------------------------------------
extern "C" void kernel_launch(void* const* d_in, const int* in_sizes, int n_in,
                              void* d_out, int out_size, void* d_ws, size_t ws_size,
                              hipStream_t stream) {
    const float* boxes  = (const float*)d_in[0];   // (512,4) f32
    const float* scores = (const float*)d_in[1];   // (512,)  f32
    const int*   labels = (const int*)  d_in[2];   // (512,)  i32
    const float* masks  = (const float*)d_in[3];   // (512,1,512,512) f32
    float* out = (float*)d_out;

    int*   ws_order = (int*)d_ws;
    float* ws_keep  = (float*)((char*)d_ws + N_BOX * sizeof(int));

    nms_sort_kernel<<<1, 512, 0, stream>>>(boxes, scores, labels, out,
                                           ws_order, ws_keep);
    mask_gather_kernel<<<dim3(32, 512), 256, 0, stream>>>(masks, out + MASK_OFF,
                                                          ws_order, ws_keep);
}